// MASGNN_69861938036807
// MI455X (gfx1250) — compile-verified
//
#include <hip/hip_runtime.h>
#include <hip/hip_bf16.h>
#include <math.h>

// ---------------------------------------------------------------------------
// MASGNN layer for MI455X (gfx1250, wave32).
//
// Restructured:
//   HS  = hidden @ Ws^T          (WMMA f32 16x16x4 GEMM, 200000x128x128)
//   HR  = rela_embed @ Wr^T      (tiny, scalar kernel: 1005x128)
//   KGB[ij] = Wkg @ concat(kg_table[i], kg_table[j]) + bkg   (4 vectors only!)
//   per edge: pre = HS[sub] + HR[rel] + KGB[ij]; alpha = sigmoid(relu(pre)·wa + ba)
//             atomicAdd(agg[obj], alpha*(hidden[sub]+rela[rel]))
//   out = agg @ Wh^T             (WMMA f32 16x16x4 GEMM)
//
// GEMM: one wave computes a full 16x128 output strip (8 accumulators); the A
// fragment is loaded once per k-step and reused across all 8 N-tiles.
// B fragments are loaded as a batch (distinct registers) *before* the 8 WMMAs
// so the scheduler can clause the loads and stagger s_wait_loadcnt, instead of
// a load->wait(0)->wmma serial chain.
// ---------------------------------------------------------------------------

typedef __attribute__((ext_vector_type(2))) float v2f;
typedef __attribute__((ext_vector_type(8))) float v8f;

#define DIMC 128   // IN_DIM == ATTN_DIM == OUT_DIM == 128

__global__ void masgnn_zero_kernel(float* __restrict__ p, long n) {
  long i = (long)blockIdx.x * blockDim.x + threadIdx.x;
  if (i < n) p[i] = 0.0f;
}

// HR[r][j] = sum_k rela[r][k] * Wr[j][k]
__global__ void masgnn_relproj_kernel(const float* __restrict__ rela,
                                      const float* __restrict__ Wr,
                                      float* __restrict__ HR, int nrel) {
  int idx = blockIdx.x * blockDim.x + threadIdx.x;
  if (idx >= nrel * DIMC) return;
  int r = idx >> 7, j = idx & 127;
  const float* a = rela + (size_t)r * DIMC;
  const float* w = Wr + (size_t)j * DIMC;
  float s = 0.0f;
  #pragma unroll 4
  for (int k = 0; k < DIMC; ++k) s = fmaf(a[k], w[k], s);
  HR[idx] = s;
}

// KGB[(i*2+t)][j] = Wkg[j][0:128]·kg[i] + Wkg[j][128:256]·kg[t] + bkg[j]
__global__ void masgnn_kgb_kernel(const float* __restrict__ kg,
                                  const float* __restrict__ Wkg,
                                  const float* __restrict__ bkg,
                                  float* __restrict__ KGB) {
  int idx = blockIdx.x * blockDim.x + threadIdx.x;
  if (idx >= 4 * DIMC) return;
  int ij = idx >> 7, j = idx & 127;
  int i = ij >> 1, t = ij & 1;
  const float* w = Wkg + (size_t)j * (2 * DIMC);
  float s = bkg[j];
  for (int c = 0; c < DIMC; ++c) s = fmaf(w[c],        kg[i * DIMC + c], s);
  for (int c = 0; c < DIMC; ++c) s = fmaf(w[DIMC + c], kg[t * DIMC + c], s);
  KGB[idx] = s;
}

// C[M x 128] = A[M x 128] @ W[128 x 128]^T using V_WMMA_F32_16X16X4_F32.
// One wave computes a 16x128 strip: 8 accumulators, A fragment reused 8x.
// A frag (16x4 f32): lane<16 row M=l16 holds K={k,k+1}; lane>=16 holds K={k+2,k+3}.
// B frag (4x16 f32): B[k][n] = W[n][k] -> contiguous float2 from W row (n-tile*16+l16).
// C/D (16x16 f32): VGPR r, lane<16 -> row m0+r, lane>=16 -> row m0+8+r.
__global__ void masgnn_gemm_awt_wmma(const float* __restrict__ A,
                                     const float* __restrict__ W,
                                     float* __restrict__ C,
                                     int M) {
  int wave = (int)((blockIdx.x * blockDim.x + threadIdx.x) >> 5);
  int lane = threadIdx.x & 31;
  int m0 = wave << 4;
  if (m0 >= M) return;                 // wave-uniform: EXEC stays all-ones
  int half = lane >> 4;                // 0 or 1
  int l16  = lane & 15;

  const float* arow  = A + (size_t)(m0 + l16) * DIMC + half * 2;
  const float* wbase = W + (size_t)l16 * DIMC + half * 2;   // n-tile 0; n-tile stride 16*DIMC

  v8f acc[8] = {};
  #pragma unroll 2
  for (int k = 0; k < DIMC; k += 4) {
    // batch all loads for this k-step into distinct registers first
    v2f a = *(const v2f*)(arow + k);
    v2f b[8];
    #pragma unroll
    for (int n = 0; n < 8; ++n) {
      b[n] = *(const v2f*)(wbase + (size_t)n * (16 * DIMC) + k);
    }
    // then consume: waits can count down (0x7, 0x6, ...) instead of 0x0 each
    #pragma unroll
    for (int n = 0; n < 8; ++n) {
      acc[n] = __builtin_amdgcn_wmma_f32_16x16x4_f32(
          /*neg_a=*/false, a, /*neg_b=*/false, b[n],
          /*c_mod=*/(short)0, acc[n], /*reuse_a=*/false, /*reuse_b=*/false);
    }
  }

  float* crow = C + (size_t)(m0 + half * 8) * DIMC + l16;
  #pragma unroll
  for (int n = 0; n < 8; ++n) {
    #pragma unroll
    for (int r = 0; r < 8; ++r) {
      crow[(size_t)r * DIMC + n * 16] = acc[n][r];
    }
  }
}

// One wave32 per edge: 128 features = 4 per lane.
__global__ void masgnn_edge_kernel(const float* __restrict__ hidden,
                                   const float* __restrict__ rela,
                                   const float* __restrict__ HS,
                                   const float* __restrict__ HR,
                                   const float* __restrict__ KGB,
                                   const float* __restrict__ wa,
                                   const float* __restrict__ ba,
                                   const int* __restrict__ edges,
                                   const int* __restrict__ d_left,
                                   float* __restrict__ agg,
                                   int n_edge) {
  int wave = (int)((blockIdx.x * blockDim.x + threadIdx.x) >> 5);
  int lane = threadIdx.x & 31;
  if (wave >= n_edge) return;          // wave-uniform

  const int left_num = d_left[0];
  const int* e = edges + (size_t)wave * 6;
  int head = e[1], rel = e[2], tail = e[3], sub = e[4], obj = e[5];
  int ij = ((head >= left_num) ? 2 : 0) + ((tail >= left_num) ? 1 : 0);

  int j = lane << 2;                   // feature offset, 16B aligned
  float4 hsw = *(const float4*)(HS  + (size_t)sub * DIMC + j);
  float4 hrw = *(const float4*)(HR  + (size_t)rel * DIMC + j);
  float4 kgw = *(const float4*)(KGB + (size_t)ij  * DIMC + j);
  float4 waw = *(const float4*)(wa + j);

  float4 pre;
  pre.x = hsw.x + hrw.x + kgw.x;
  pre.y = hsw.y + hrw.y + kgw.y;
  pre.z = hsw.z + hrw.z + kgw.z;
  pre.w = hsw.w + hrw.w + kgw.w;

  float t = fmaxf(pre.x, 0.0f) * waw.x + fmaxf(pre.y, 0.0f) * waw.y +
            fmaxf(pre.z, 0.0f) * waw.z + fmaxf(pre.w, 0.0f) * waw.w;
  // wave32 butterfly reduction -> every lane holds the full dot product
  #pragma unroll
  for (int off = 16; off > 0; off >>= 1) t += __shfl_xor(t, off, 32);

  float alpha = 1.0f / (1.0f + __expf(-(t + ba[0])));

  float4 h = *(const float4*)(hidden + (size_t)sub * DIMC + j);
  float4 r = *(const float4*)(rela   + (size_t)rel * DIMC + j);

  float* dst = agg + (size_t)obj * DIMC + j;
  atomicAdd(dst + 0, alpha * (h.x + r.x));
  atomicAdd(dst + 1, alpha * (h.y + r.y));
  atomicAdd(dst + 2, alpha * (h.z + r.z));
  atomicAdd(dst + 3, alpha * (h.w + r.w));
}

extern "C" void kernel_launch(void* const* d_in, const int* in_sizes, int n_in,
                              void* d_out, int out_size, void* d_ws, size_t ws_size,
                              hipStream_t stream) {
  const float* hidden   = (const float*)d_in[0];
  const float* rela     = (const float*)d_in[1];
  const float* kg_table = (const float*)d_in[2];
  const float* Ws       = (const float*)d_in[3];
  const float* Wr       = (const float*)d_in[4];
  const float* Wkg      = (const float*)d_in[5];
  const float* bkg      = (const float*)d_in[6];
  const float* wa       = (const float*)d_in[7];
  const float* ba       = (const float*)d_in[8];
  const float* Wh       = (const float*)d_in[9];
  const int*   edges    = (const int*)d_in[10];
  const int*   d_left   = (const int*)d_in[12];

  const int n_node = in_sizes[0] / DIMC;        // 200000 (divisible by 16)
  const int n_rel  = in_sizes[1] / DIMC;        // 1005
  const int n_edge = in_sizes[10] / 6;          // 1000000

  // workspace layout (all 256B aligned)
  float* agg = (float*)d_ws;                               // n_node*128 f32
  float* HS  = agg + (size_t)n_node * DIMC;                // n_node*128 f32
  float* HR  = HS  + (size_t)n_node * DIMC;                // n_rel*128 f32
  float* KGB = HR  + (size_t)n_rel * DIMC;                 // 4*128 f32

  // 1) zero the aggregation buffer (workspace is not re-poisoned between runs)
  long aggN = (long)n_node * DIMC;
  masgnn_zero_kernel<<<(int)((aggN + 255) / 256), 256, 0, stream>>>(agg, aggN);

  // 2) tiny precomputes
  masgnn_relproj_kernel<<<(n_rel * DIMC + 255) / 256, 256, 0, stream>>>(rela, Wr, HR, n_rel);
  masgnn_kgb_kernel<<<2, 256, 0, stream>>>(kg_table, Wkg, bkg, KGB);

  // 3) HS = hidden @ Ws^T via WMMA (one wave per 16x128 strip, 8 waves/block)
  int strips = n_node / 16;                      // 12500
  int gemm_blocks = (strips + 7) / 8;            // 1563
  masgnn_gemm_awt_wmma<<<gemm_blocks, 256, 0, stream>>>(hidden, Ws, HS, n_node);

  // 4) per-edge attention + scatter-add (one wave per edge)
  masgnn_edge_kernel<<<(n_edge + 7) / 8, 256, 0, stream>>>(
      hidden, rela, HS, HR, KGB, wa, ba, edges, d_left, agg, n_edge);

  // 5) out = agg @ Wh^T via WMMA
  masgnn_gemm_awt_wmma<<<gemm_blocks, 256, 0, stream>>>(agg, Wh, (float*)d_out, n_node);
}